// MyModel_13950053777458
// MI455X (gfx1250) — compile-verified
//
#include <hip/hip_runtime.h>

typedef __attribute__((ext_vector_type(2))) float v2f;
typedef __attribute__((ext_vector_type(4))) float v4f;
typedef __attribute__((ext_vector_type(8))) float v8f;

// One 16-row slab: H(16x2) = [x0 x1 1 0](16x4) x [W^T; b; 0](4x16) via
// V_WMMA_F32_16X16X4_F32, then out[m] = g * (H[m][0]^2 + H[m][1]^2).
// D layout: VGPR r -> M=r (lanes 0-15, N=lane), M=r+8 (lanes 16-31, N=lane-16).
// After squaring + shfl_xor(1), lane 0 holds rows 0..7, lane 16 holds rows 8..15
// in 8 consecutive VGPRs -> two b128 stores each.
__device__ __forceinline__ void qexp_tile16(float ax, float ay,
                                            float bv0, float bv1,
                                            float g, float* __restrict__ outp,
                                            int lane) {
  v2f a; a.x = ax; a.y = ay;        // A: VGPR0=K0/K2, VGPR1=K1/K3
  v2f b; b.x = bv0; b.y = bv1;      // B: VGPR0=rows K0,K2; VGPR1=rows K1,K3
  v8f c = {};
  v8f d = __builtin_amdgcn_wmma_f32_16x16x4_f32(
      /*neg_a=*/false, a, /*neg_b=*/false, b,
      /*c_mod=*/(short)0, c, /*reuse_a=*/false, /*reuse_b=*/false);

  float t0, t1, t2, t3, t4, t5, t6, t7;
  {
    float e;
    e = d[0] * d[0]; t0 = g * (e + __shfl_xor(e, 1));
    e = d[1] * d[1]; t1 = g * (e + __shfl_xor(e, 1));
    e = d[2] * d[2]; t2 = g * (e + __shfl_xor(e, 1));
    e = d[3] * d[3]; t3 = g * (e + __shfl_xor(e, 1));
    e = d[4] * d[4]; t4 = g * (e + __shfl_xor(e, 1));
    e = d[5] * d[5]; t5 = g * (e + __shfl_xor(e, 1));
    e = d[6] * d[6]; t6 = g * (e + __shfl_xor(e, 1));
    e = d[7] * d[7]; t7 = g * (e + __shfl_xor(e, 1));
  }
  if ((lane & 15) == 0) {                 // lanes 0 and 16
    float* p = outp + ((lane >> 4) << 3); // rows 0..7 or 8..15
    v4f lo = {t0, t1, t2, t3};
    v4f hi = {t4, t5, t6, t7};
    *(v4f*)p = lo;
    *(v4f*)(p + 4) = hi;
  }
}

__global__ void __launch_bounds__(256)
qexp_rx_kernel(const float* __restrict__ x, const float* __restrict__ theta,
               const float* __restrict__ W, const float* __restrict__ bias,
               float* __restrict__ out, int n) {
  const int lane = threadIdx.x & 31;
  const int waveInBlk = threadIdx.x >> 5;
  const int wavesPerBlk = blockDim.x >> 5;
  const long long waveId = (long long)blockIdx.x * wavesPerBlk + waveInBlk;
  const long long waveStride = (long long)gridDim.x * wavesPerBlk;

  const float th = theta[0];
  const float cv = cosf(0.5f * th);
  const float sv = sinf(0.5f * th);
  const float g = cv * cv + sv * sv;  // faithful (c^2+s^2) scaling, ~1.0

  const float w00 = W[0], w01 = W[1], w10 = W[2], w11 = W[3];
  const float b0 = bias[0], b1 = bias[1];

  // B matrix (4x16): B[k][n] = W[n][k] for k=0,1; B[2][n] = b[n]; B[3][*]=0.
  float bv0 = 0.0f, bv1 = 0.0f;
  if (lane == 0)       { bv0 = w00; bv1 = w01; }
  else if (lane == 1)  { bv0 = w10; bv1 = w11; }
  else if (lane == 16) { bv0 = b0; }
  else if (lane == 17) { bv0 = b1; }

  const long long ntiles = (long long)n >> 5;  // 32 rows per wave-tile
  const bool loHalf = lane < 16;

  for (long long tile = waveId; tile < ntiles; tile += waveStride) {
    const long long row = tile << 5;

    // Coalesced 256B wave load: row (row+lane), 8B per lane.
    const v2f xv = *(const v2f*)(x + ((row + lane) << 1));

    // CDNA5 prefetch of this wave's next tile (speculative; OOB dropped).
    __builtin_prefetch(x + (((tile + waveStride) << 5) << 1), 0, 1);

    // Rows 16..31 data lives in lanes 16..31; swap halves for second WMMA.
    const float y0 = __shfl_xor(xv.x, 16);
    const float y1 = __shfl_xor(xv.y, 16);

    // Augmented A column: lanes>=16 supply the K=2 "ones" (bias) / K=3 zeros.
    qexp_tile16(loHalf ? xv.x : 1.0f, loHalf ? xv.y : 0.0f,
                bv0, bv1, g, out + row, lane);
    qexp_tile16(loHalf ? y0 : 1.0f, loHalf ? y1 : 0.0f,
                bv0, bv1, g, out + row + 16, lane);
  }

  // Scalar tail for n % 32 (empty for B = 2^23, kept for generality).
  const long long tailStart = ntiles << 5;
  for (long long i = tailStart + (long long)blockIdx.x * blockDim.x + threadIdx.x;
       i < (long long)n;
       i += (long long)gridDim.x * blockDim.x) {
    const float xx0 = x[2 * i], xx1 = x[2 * i + 1];
    const float h0 = fmaf(w01, xx1, fmaf(w00, xx0, b0));
    const float h1 = fmaf(w11, xx1, fmaf(w10, xx0, b1));
    out[i] = g * (h0 * h0 + h1 * h1);
  }
}

extern "C" void kernel_launch(void* const* d_in, const int* in_sizes, int n_in,
                              void* d_out, int out_size, void* d_ws, size_t ws_size,
                              hipStream_t stream) {
  const float* x     = (const float*)d_in[0];
  const float* theta = (const float*)d_in[1];
  const float* W     = (const float*)d_in[2];
  const float* bias  = (const float*)d_in[3];
  float* out = (float*)d_out;

  const int n = in_sizes[0] / 2;  // number of rows (B)

  const int threads = 256;                       // 8 wave32 per block
  long long wavesNeeded = ((long long)n + 31) / 32;
  long long blocksLL = (wavesNeeded + 7) / 8;
  if (blocksLL < 1) blocksLL = 1;
  if (blocksLL > 2048) blocksLL = 2048;          // persistent grid-stride
  const int blocks = (int)blocksLL;

  qexp_rx_kernel<<<blocks, threads, 0, stream>>>(x, theta, W, bias, out, n);
}